// PGA_Conv_comp_grad_10900626997254
// MI455X (gfx1250) — compile-verified
//
#include <hip/hip_runtime.h>

typedef float v2f __attribute__((ext_vector_type(2)));
typedef float v8f __attribute__((ext_vector_type(8)));

namespace {
constexpr int kK   = 4;
constexpr int kTS  = 128;
constexpr int kNT  = 64;
constexpr int kM   = 8;
constexpr int kNRF = 8;
constexpr float kSigma2 = 1.0f;
constexpr float kEps    = 1.0e-4f;
constexpr float kLog2   = 0.6931471805599453f;   // ln(2)

// float offsets into d_out (complex64 stored as interleaved re/im floats)
constexpr size_t OFF_FRAD = (size_t)kK * kTS * kNT * kNRF * 2;           // 524288
constexpr size_t OFF_WCOM = OFF_FRAD + (size_t)kTS * kNT * kNRF * 2;     // 655360
constexpr size_t OFF_WRAD = OFF_WCOM + (size_t)kK * kTS * kNRF * kM * 2; // 720896
}

static __device__ __forceinline__ v8f wmma4(v2f a, v2f b, v8f c) {
  // D = A(16x4,f32) x B(4x16,f32) + C(16x16,f32); f32 variant has no A/B NEG.
  return __builtin_amdgcn_wmma_f32_16x16x4_f32(false, a, false, b, (short)0, c,
                                               false, false);
}

__global__ __launch_bounds__(128) void pga_grads_kernel(
    const float* __restrict__ gHre, const float* __restrict__ gHim,
    const float* __restrict__ gFre, const float* __restrict__ gFim,
    const float* __restrict__ gWre, const float* __restrict__ gWim,
    const float* __restrict__ gRre, const float* __restrict__ gRim,
    float* __restrict__ out) {
  const int ts   = blockIdx.x;          // one workgroup per TS slice
  const int t    = threadIdx.x;         // 0..127 (4 wave32)
  const int lane = t & 31;
  const int wave = t >> 5;              // 0..3
  const int l16  = lane & 15;           // N index (B/C/D) or M index (A)
  const int hk   = lane >> 4;           // half of the wave

  __shared__ float sFre[64][8],  sFim[64][8];
  __shared__ float sHre[64][8],  sHim[64][8];
  __shared__ float sWre[8][8],   sWim[8][8];
  __shared__ float sVre[8][8],   sVim[8][8];
  __shared__ float sGpk[64][16];                 // [i][0..7]=G_re, [i][8..15]=G_im
  __shared__ float sEre[64][64], sEim[64][64];
  __shared__ float sPre[8][64],  sPim[8][64];    // P = F^H E
  __shared__ float sXaccRe[64][8], sXaccIm[64][8];   // sum_k (E G)
  __shared__ float sgFre[64][8],   sgFim[64][8];     // sum_{k,m} h_m r_m
  __shared__ float sAre[8][8],   sAim[8][8];     // a_m = F^H h_m  ([m][r])
  __shared__ float sRRre[8][8],  sRRim[8][8];    // r_row per m ([m][j])
  __shared__ float sZFre[8],     sZFim[8];       // z_m/(d1+eps)
  __shared__ float sQre[8][8],   sQim[8][8];     // Q = P F

  for (int idx = t; idx < 512; idx += 128) {
    sXaccRe[idx >> 3][idx & 7] = 0.0f;
    sXaccIm[idx >> 3][idx & 7] = 0.0f;
    sgFre[idx >> 3][idx & 7]   = 0.0f;
    sgFim[idx >> 3][idx & 7]   = 0.0f;
  }
  __syncthreads();

  for (int k = 0; k < kK; ++k) {
    const size_t slice = (size_t)(k * kTS + ts);
    // ---- stage inputs into LDS; prefetch the 64x64 R tile ----------------
    __builtin_prefetch(gRre + slice * 4096 + t * 32, 0, 1);
    __builtin_prefetch(gRim + slice * 4096 + t * 32, 0, 1);
    const size_t b512 = slice * 512;
    for (int i = t; i < 512; i += 128) {
      sFre[i >> 3][i & 7] = gFre[b512 + i];
      sFim[i >> 3][i & 7] = gFim[b512 + i];
      sHre[i >> 3][i & 7] = gHre[b512 + i];
      sHim[i >> 3][i & 7] = gHim[b512 + i];
    }
    if (t < 64) {
      sWre[t >> 3][t & 7] = gWre[slice * 64 + t];
      sWim[t >> 3][t & 7] = gWim[slice * 64 + t];
    }
    __syncthreads();

    // ---- V = W W^H (8x8) -------------------------------------------------
    if (t < 64) {
      const int i = t >> 3, j = t & 7;
      float vr = 0.0f, vi = 0.0f;
      #pragma unroll
      for (int r = 0; r < 8; ++r) {
        const float ar = sWre[i][r], ai = sWim[i][r];
        const float br = sWre[j][r], bi = sWim[j][r];
        vr += ar * br + ai * bi;     // W[i][r]*conj(W[j][r])
        vi += ai * br - ar * bi;
      }
      sVre[i][j] = vr; sVim[i][j] = vi;
    }
    __syncthreads();

    // ---- G = F V (64x8), packed [re|im] ----------------------------------
    for (int idx = t; idx < 512; idx += 128) {
      const int i = idx >> 3, j = idx & 7;
      float gr = 0.0f, gi = 0.0f;
      #pragma unroll
      for (int r = 0; r < 8; ++r) {
        const float fr = sFre[i][r], fi = sFim[i][r];
        const float vr = sVre[r][j], vi = sVim[r][j];
        gr += fr * vr - fi * vi;
        gi += fr * vi + fi * vr;
      }
      sGpk[i][j] = gr; sGpk[i][j + 8] = gi;
    }
    __syncthreads();

    // ---- E = G F^H - R (WMMA, 64x64); wave owns row tile ti=wave ---------
    {
      const int ti = wave;
      #pragma unroll
      for (int tj = 0; tj < 4; ++tj) {
        v8f accre = {}; v8f accim = {};
        #pragma unroll
        for (int kb = 0; kb < 8; kb += 4) {
          const int kk = kb + 2 * hk;
          const v2f aGre = { sGpk[16 * ti + l16][kk],     sGpk[16 * ti + l16][kk + 1] };
          const v2f aGim = { sGpk[16 * ti + l16][8 + kk], sGpk[16 * ti + l16][8 + kk + 1] };
          const v2f bFre = { sFre[16 * tj + l16][kk],     sFre[16 * tj + l16][kk + 1] };
          const v2f bFim = { sFim[16 * tj + l16][kk],     sFim[16 * tj + l16][kk + 1] };
          const v2f bFimN = { -bFim.x, -bFim.y };
          accre = wmma4(aGre, bFre, accre);   // +Gre*Fre^T
          accre = wmma4(aGim, bFim, accre);   // +Gim*Fim^T
          accim = wmma4(aGim, bFre, accim);   // +Gim*Fre^T
          accim = wmma4(aGre, bFimN, accim);  // -Gre*Fim^T
        }
        #pragma unroll
        for (int v = 0; v < 8; ++v) {
          const int i = 16 * ti + v + 8 * hk;
          const int j = 16 * tj + l16;
          sEre[i][j] = accre[v] - gRre[slice * 4096 + i * 64 + j];
          sEim[i][j] = accim[v] - gRim[slice * 4096 + i * 64 + j];
        }
      }
    }
    __syncthreads();

    // ---- X = E G (64x8) via packed B=[Gre|Gim]; accumulate over k --------
    {
      const int ti = wave;
      v8f accA = {}; v8f accB = {};
      #pragma unroll
      for (int kb = 0; kb < 64; kb += 4) {
        const int kk = kb + 2 * hk;
        const v2f aEre = { sEre[16 * ti + l16][kk], sEre[16 * ti + l16][kk + 1] };
        const v2f aEim = { sEim[16 * ti + l16][kk], sEim[16 * ti + l16][kk + 1] };
        const v2f bG   = { sGpk[kk][l16],           sGpk[kk + 1][l16] };
        accA = wmma4(aEre, bG, accA);   // [Ere*Gre | Ere*Gim]
        accB = wmma4(aEim, bG, accB);   // [Eim*Gre | Eim*Gim]
      }
      #pragma unroll
      for (int v = 0; v < 8; ++v) {
        const float a_hi = __shfl(accA[v], lane + 8, 32);  // col j+8, same rows
        const float b_hi = __shfl(accB[v], lane + 8, 32);
        if (l16 < 8) {
          const int i = 16 * ti + v + 8 * hk;
          const int j = l16;
          sXaccRe[i][j] += accA[v] - b_hi;   // Ere Gre - Eim Gim
          sXaccIm[i][j] += a_hi + accB[v];   // Ere Gim + Eim Gre
        }
      }
    }

    // ---- P = F^H E (8x64) via packed A=[Fre^T;Fim^T]; wave owns col tile -
    {
      const int tj = wave;
      v8f accR = {}; v8f accI = {};
      #pragma unroll
      for (int kb = 0; kb < 64; kb += 4) {
        const int kk = kb + 2 * hk;
        const float (*srcA)[8] = (l16 < 8) ? sFre : sFim;
        const int ca = l16 & 7;
        const v2f aF   = { srcA[kk][ca],               srcA[kk + 1][ca] };
        const v2f bEre = { sEre[kk][16 * tj + l16],    sEre[kk + 1][16 * tj + l16] };
        const v2f bEim = { sEim[kk][16 * tj + l16],    sEim[kk + 1][16 * tj + l16] };
        accR = wmma4(aF, bEre, accR);   // [Fre^T Ere ; Fim^T Ere]
        accI = wmma4(aF, bEim, accI);   // [Fre^T Eim ; Fim^T Eim]
      }
      #pragma unroll
      for (int v = 0; v < 8; ++v) {
        const float r_hi = __shfl(accR[v], lane + 16, 32);  // rows 8..15
        const float i_hi = __shfl(accI[v], lane + 16, 32);
        if (hk == 0) {
          const int c = 16 * tj + l16;
          sPre[v][c] = accR[v] + i_hi;   // Fre^T Ere + Fim^T Eim
          sPim[v][c] = accI[v] - r_hi;   // Fre^T Eim - Fim^T Ere
        }
      }
    }
    __syncthreads();

    // ---- grad_W_rad = 2 (P F) W / K --------------------------------------
    if (t < 64) {
      const int r = t >> 3, s = t & 7;
      float qr = 0.0f, qi = 0.0f;
      for (int i = 0; i < 64; ++i) {
        const float pr = sPre[r][i], pi = sPim[r][i];
        const float fr = sFre[i][s], fi = sFim[i][s];
        qr += pr * fr - pi * fi;
        qi += pr * fi + pi * fr;
      }
      sQre[r][s] = qr; sQim[r][s] = qi;
    }
    __syncthreads();
    if (t < 64) {
      const int r = t >> 3, m = t & 7;
      float trr = 0.0f, tri = 0.0f;
      #pragma unroll
      for (int s = 0; s < 8; ++s) {
        const float qr = sQre[r][s], qi = sQim[r][s];
        const float wr = sWre[s][m], wi = sWim[s][m];
        trr += qr * wr - qi * wi;
        tri += qr * wi + qi * wr;
      }
      const size_t o = OFF_WRAD + (slice * 64 + r * 8 + m) * 2;
      out[o]     = 2.0f * trr / kK;
      out[o + 1] = 2.0f * tri / kK;
    }

    // ---- a_m = F^H h_m ----------------------------------------------------
    if (t < 64) {
      const int m = t >> 3, r = t & 7;
      float ar = 0.0f, ai = 0.0f;
      for (int i = 0; i < 64; ++i) {
        const float fr = sFre[i][r], fi = sFim[i][r];
        const float hr = sHre[i][m], hi = sHim[i][m];
        ar += fr * hr + fi * hi;     // conj(F)*H
        ai += fr * hi - fi * hr;
      }
      sAre[m][r] = ar; sAim[m][r] = ai;
    }
    __syncthreads();

    // ---- per-m scalars: q=a^H V, s1=q.a, z=a^H w_m, r_row ----------------
    if (t < 8) {
      const int m = t;
      float qre[8], qim[8];
      #pragma unroll
      for (int j = 0; j < 8; ++j) {
        float qr = 0.0f, qi = 0.0f;
        #pragma unroll
        for (int r = 0; r < 8; ++r) {
          const float ar = sAre[m][r], ai = sAim[m][r];
          const float vr = sVre[r][j], vi = sVim[r][j];
          qr += ar * vr + ai * vi;   // conj(a)*V
          qi += ar * vi - ai * vr;
        }
        qre[j] = qr; qim[j] = qi;
      }
      float s1r = 0.0f, s1i = 0.0f;
      #pragma unroll
      for (int j = 0; j < 8; ++j) {
        s1r += qre[j] * sAre[m][j] - qim[j] * sAim[m][j];
        s1i += qre[j] * sAim[m][j] + qim[j] * sAre[m][j];
      }
      float zr = 0.0f, zi = 0.0f;
      #pragma unroll
      for (int r = 0; r < 8; ++r) {
        const float ar = sAre[m][r], ai = sAim[m][r];
        const float wr = sWre[r][m], wi = sWim[r][m];
        zr += ar * wr + ai * wi;     // conj(a)*w
        zi += ar * wi - ai * wr;
      }
      const float d1r = kLog2 * (s1r + kSigma2) + kEps, d1i = kLog2 * s1i;
      const float zz  = zr * zr + zi * zi;
      const float d2r = kLog2 * (s1r - zz + kSigma2) + kEps, d2i = kLog2 * s1i;
      const float n1 = d1r * d1r + d1i * d1i;
      const float i1r = d1r / n1, i1i = -d1i / n1;
      const float n2 = d2r * d2r + d2i * d2i;
      const float i2r = d2r / n2, i2i = -d2i / n2;
      #pragma unroll
      for (int j = 0; j < 8; ++j) {
        const float wjr = sWre[j][m], wji = sWim[j][m];
        const float zcr = zr * wjr + zi * wji;   // z*conj(W[j][m])
        const float zci = zi * wjr - zr * wji;
        const float ur = qre[j] - zcr, ui = qim[j] - zci;
        const float t1r = qre[j] * i1r - qim[j] * i1i;
        const float t1i = qre[j] * i1i + qim[j] * i1r;
        const float t2r = ur * i2r - ui * i2i;
        const float t2i = ur * i2i + ui * i2r;
        sRRre[m][j] = t1r - t2r;
        sRRim[m][j] = t1i - t2i;
      }
      sZFre[m] = zr * i1r - zi * i1i;
      sZFim[m] = zr * i1i + zi * i1r;
    }
    __syncthreads();

    // ---- grad_W_com[r][m] = a_m[r] * z_m/(d1+eps) / K --------------------
    if (t < 64) {
      const int r = t >> 3, m = t & 7;
      const float ar = sAre[m][r], ai = sAim[m][r];
      const size_t o = OFF_WCOM + (slice * 64 + r * 8 + m) * 2;
      out[o]     = (ar * sZFre[m] - ai * sZFim[m]) / kK;
      out[o + 1] = (ar * sZFim[m] + ai * sZFre[m]) / kK;
    }

    // ---- gF accumulation: gF += H(64x8) @ r_row(8x8) ---------------------
    for (int idx = t; idx < 512; idx += 128) {
      const int i = idx >> 3, j = idx & 7;
      float gr = 0.0f, gi = 0.0f;
      #pragma unroll
      for (int m = 0; m < 8; ++m) {
        const float hr = sHre[i][m], hi = sHim[i][m];
        const float rr = sRRre[m][j], ri = sRRim[m][j];
        gr += hr * rr - hi * ri;
        gi += hr * ri + hi * rr;
      }
      sgFre[i][j] += gr;
      sgFim[i][j] += gi;
    }
    __syncthreads();
  }  // k loop

  // ---- write grad_F_rad (TS,NT,NRF) and grad_F_com (repeated over K) -----
  for (int idx = t; idx < 512; idx += 128) {
    const size_t orad = OFF_FRAD + ((size_t)ts * 512 + idx) * 2;
    out[orad]     = 2.0f * sXaccRe[idx >> 3][idx & 7] / kK;
    out[orad + 1] = 2.0f * sXaccIm[idx >> 3][idx & 7] / kK;
    const float gr = sgFre[idx >> 3][idx & 7] / kK;
    const float gi = sgFim[idx >> 3][idx & 7] / kK;
    #pragma unroll
    for (int kk = 0; kk < kK; ++kk) {
      const size_t oc = (((size_t)(kk * kTS + ts)) * 512 + idx) * 2;
      out[oc]     = gr;
      out[oc + 1] = gi;
    }
  }
}

extern "C" void kernel_launch(void* const* d_in, const int* in_sizes, int n_in,
                              void* d_out, int out_size, void* d_ws, size_t ws_size,
                              hipStream_t stream) {
  (void)in_sizes; (void)n_in; (void)out_size; (void)d_ws; (void)ws_size;
  const float* Hre = (const float*)d_in[0];
  const float* Him = (const float*)d_in[1];
  const float* Fre = (const float*)d_in[2];
  const float* Fim = (const float*)d_in[3];
  const float* Wre = (const float*)d_in[4];
  const float* Wim = (const float*)d_in[5];
  const float* Rre = (const float*)d_in[6];
  const float* Rim = (const float*)d_in[7];
  pga_grads_kernel<<<dim3(kTS), dim3(128), 0, stream>>>(
      Hre, Him, Fre, Fim, Wre, Wim, Rre, Rim, (float*)d_out);
}